// FCOSPostProcessor_35665408426317
// MI455X (gfx1250) — compile-verified
//
#include <hip/hip_runtime.h>
#include <hip/hip_bf16.h>
#include <stdint.h>

// ---------------- problem constants ----------------
#define NLEV 5
#define NLOC 21824           // 16384+4096+1024+256+64
#define NCLS 80
#define BATCH 16
#define CHUNK 64             // locations per workgroup tile (divides every level size)
#define CHUNK_SC (CHUNK*NCLS)  // 5120 scores per tile
#define NCHUNK (NLOC/CHUNK)  // 341
#define TOPN 1000
#define MPER (NLEV*TOPN)     // 5000 candidates per batch entering NMS
#define MPAD 5008
#define SORTN 8192           // bitonic pad for per-batch sort
#define CANDCAP 4096         // per-(b,l) collection cap (bitonic pad too)
#define NBIN 1024
#define PRE_THR 0.05f
#define THR_LOGIT (-2.9444389792f)  // logit(0.05)
#define NMS_THR 0.6f
#define POSTN 100

__device__ __constant__ int kLevEnd[NLEV] = {16384, 20480, 21504, 21760, 21824};

static __device__ __forceinline__ float sigmoidf_(float x) {
    return 1.0f / (1.0f + __expf(-x));
}
static __device__ __forceinline__ int lev_of(int loc) {
    int l = 0;
#pragma unroll
    for (int i = 0; i < NLEV - 1; ++i) if (loc >= kLevEnd[i]) l = i + 1;
    return l;
}

// ---------------- CDNA5 feature guards ----------------
#if defined(__gfx1250__) && __has_builtin(__builtin_amdgcn_tensor_load_to_lds) && \
    __has_builtin(__builtin_amdgcn_s_wait_tensorcnt)
#define HAVE_TDM 1
typedef unsigned int u32x4 __attribute__((ext_vector_type(4)));
typedef int          i32x4 __attribute__((ext_vector_type(4)));
typedef int          i32x8 __attribute__((ext_vector_type(8)));
#endif

typedef float v2f __attribute__((ext_vector_type(2)));
typedef float v8f __attribute__((ext_vector_type(8)));
#if defined(__gfx1250__) && __has_builtin(__builtin_amdgcn_wmma_f32_16x16x4_f32)
#define HAVE_WMMA4 1
#endif

// ============================================================
// Kernel 1: TDM-staged scoring + per-(b,level) weighted-score histogram
// grid = BATCH*NCHUNK blocks x 256 threads
// ============================================================
__global__ __launch_bounds__(256) void k_hist(
    const float* __restrict__ cls, const float* __restrict__ ctr,
    unsigned int* __restrict__ gh)
{
    __shared__ float tile[CHUNK_SC];       // 20KB cls tile (TDM destination)
    __shared__ float ctrs[CHUNK];
    __shared__ unsigned int lh[NBIN];
    const int tid  = threadIdx.x;
    const int b    = blockIdx.x / NCHUNK;
    const int ch   = blockIdx.x % NCHUNK;
    const int loc0 = ch * CHUNK;
    const float* src = cls + ((size_t)b * NLOC + loc0) * NCLS;

    for (int i = tid; i < NBIN; i += 256) lh[i] = 0u;
    if (tid < CHUNK) ctrs[tid] = sigmoidf_(ctr[(size_t)b * NLOC + loc0 + tid]);

#if HAVE_TDM
    // One TDM descriptor per workgroup: 1-D tile of CHUNK_SC f32 -> LDS.
    if (tid < 32) {  // wave 0 only issues the DMA (uniform branch)
        unsigned long long ga = (unsigned long long)(uintptr_t)src;
        unsigned int lds_off  = (unsigned int)(uintptr_t)(&tile[0]); // addr[31:0] = LDS offset
        u32x4 g0;
        g0[0] = 1u;                                              // count=1, user descriptor
        g0[1] = lds_off;                                         // lds_addr
        g0[2] = (unsigned int)ga;                                // global_addr[31:0]
        g0[3] = (unsigned int)((ga >> 32) & 0x1FFFFFFull) | (2u << 30); // addr[56:32] | type=2
        i32x8 g1;
        g1[0] = 0x00020000;                                      // wg_mask=0, data_size=4B
        g1[1] = (int)((unsigned)CHUNK_SC << 16);                 // tensor_dim0 lo16
        g1[2] = (int)(((unsigned)CHUNK_SC >> 16) | (1u << 16));  // td0 hi16 | tensor_dim1 lo16
        g1[3] = (int)((unsigned)CHUNK_SC << 16);                 // td1 hi16(0) | tile_dim0
        g1[4] = 1;                                               // tile_dim1=1, tile_dim2=0
        g1[5] = CHUNK_SC;                                        // td0_stride lo32
        g1[6] = 0;
        g1[7] = 0;
        i32x4 z4 = {};
        i32x8 z8 = {};
        // clang-23 / therock-10.0 6-arg form: (g0, g1, g2, g3, g4, cpol)
        __builtin_amdgcn_tensor_load_to_lds(g0, g1, z4, z4, z8, 0);
        __builtin_amdgcn_s_wait_tensorcnt(0);
    }
#else
    for (int i = tid; i < CHUNK_SC; i += 256) tile[i] = src[i];
#endif
    __syncthreads();

    const int bl = b * NLEV + lev_of(loc0);
    for (int s = tid; s < CHUNK_SC; s += 256) {
        float v = tile[s];
        if (v > THR_LOGIT) {                       // sigmoid(v) > 0.05
            float w = sigmoidf_(v) * ctrs[s / NCLS];
            int bin = (int)(w * (float)NBIN);
            bin = bin > (NBIN - 1) ? (NBIN - 1) : (bin < 0 ? 0 : bin);
            atomicAdd(&lh[bin], 1u);
        }
    }
    __syncthreads();
    unsigned int* dst = gh + (size_t)bl * NBIN;
    for (int i = tid; i < NBIN; i += 256) {
        unsigned int c = lh[i];
        if (c) atomicAdd(&dst[i], c);
    }
}

// ============================================================
// Kernel 2: WMMA histogram reduction + threshold-bin scan
// grid = BATCH*NLEV blocks x 64 threads
// ============================================================
__global__ __launch_bounds__(64) void k_thresh(
    const unsigned int* __restrict__ gh, int* __restrict__ Tout,
    float* __restrict__ totout)
{
    const int bl  = blockIdx.x;
    const int tid = threadIdx.x;
    const unsigned int* h = gh + (size_t)bl * NBIN;
    if (tid < 32) {                       // wave 0, EXEC all-1s (WMMA requirement)
        float total;
#if HAVE_WMMA4
        // C += A(16x4 bin counts) x B(4x16 ones): exact f32 integer sums.
        // A layout per ISA: lanes 0-15 hold K=0,1 (vgpr0,1); lanes 16-31 hold K=2,3.
        const int M     = tid & 15;
        const int kbase = (tid < 16) ? 0 : 2;
        v8f acc = {};
        v2f ones; ones[0] = 1.0f; ones[1] = 1.0f;
        for (int t = 0; t < 16; ++t) {
            v2f a;
            a[0] = (float)h[t * 64 + M * 4 + kbase + 0];
            a[1] = (float)h[t * 64 + M * 4 + kbase + 1];
            acc = __builtin_amdgcn_wmma_f32_16x16x4_f32(false, a, false, ones,
                                                        (short)0, acc, false, false);
        }
        // column 0 lives on lane 0 (rows 0-7) and lane 16 (rows 8-15)
        float s8 = acc[0]+acc[1]+acc[2]+acc[3]+acc[4]+acc[5]+acc[6]+acc[7];
        total = __shfl(s8, 0, 32) + __shfl(s8, 16, 32);
#else
        unsigned int p = 0;
        for (int i = tid; i < NBIN; i += 32) p += h[i];
        float pf = (float)p;
        for (int off = 16; off; off >>= 1) pf += __shfl_down(pf, off, 32);
        total = __shfl(pf, 0, 32);
#endif
        if (tid == 0) {
            totout[bl] = total;
            int T = 0;
            if (total >= (float)TOPN) {   // WMMA total gates the suffix scan
                unsigned int acc2 = 0;
                for (int i = NBIN - 1; i >= 0; --i) {
                    acc2 += h[i];
                    if (acc2 >= TOPN) { T = i; break; }
                }
            }
            Tout[bl] = T;
        }
    }
}

// ============================================================
// Kernel 3: second pass — collect candidates with bin >= T
// grid = BATCH*NCHUNK blocks x 256 threads
// ============================================================
__global__ __launch_bounds__(256) void k_collect(
    const float* __restrict__ cls, const float* __restrict__ ctr,
    const int* __restrict__ Tin, unsigned int* __restrict__ cnt,
    float* __restrict__ candV, unsigned int* __restrict__ candI)
{
    __shared__ float ctrs[CHUNK];
    const int tid  = threadIdx.x;
    const int b    = blockIdx.x / NCHUNK;
    const int ch   = blockIdx.x % NCHUNK;
    const int loc0 = ch * CHUNK;
    const int bl   = b * NLEV + lev_of(loc0);
    const int T    = Tin[bl];
    const float* src = cls + ((size_t)b * NLOC + loc0) * NCLS;
    if (tid < CHUNK) ctrs[tid] = sigmoidf_(ctr[(size_t)b * NLOC + loc0 + tid]);
    __syncthreads();
    for (int s = tid; s < CHUNK_SC; s += 256) {
        __builtin_prefetch(&src[s + 256], 0, 1);   // global_prefetch_b8
        float v = src[s];
        if (v > THR_LOGIT) {
            float w = sigmoidf_(v) * ctrs[s / NCLS];
            int bin = (int)(w * (float)NBIN);
            bin = bin > (NBIN - 1) ? (NBIN - 1) : (bin < 0 ? 0 : bin);
            if (bin >= T) {
                unsigned int p = atomicAdd(&cnt[bl], 1u);
                if (p < CANDCAP) {
                    size_t o = (size_t)bl * CANDCAP + p;
                    candV[o] = w;
                    candI[o] = (unsigned int)((loc0 + s / NCLS) * NCLS + (s % NCLS));
                }
            }
        }
    }
}

// ============================================================
// Kernel 4: per-(b,level) bitonic top-1000 + box decode
// grid = BATCH*NLEV blocks x 512 threads
// ============================================================
__global__ __launch_bounds__(512) void k_sortlevel(
    const float* __restrict__ candV, const unsigned int* __restrict__ candI,
    const unsigned int* __restrict__ cnt,
    const float* __restrict__ reg, const float* __restrict__ locs,
    const int* __restrict__ imgsz,
    float* __restrict__ mbox, float* __restrict__ msc, unsigned int* __restrict__ mcls)
{
    __shared__ float        key[CANDCAP];
    __shared__ unsigned int val[CANDCAP];
    const int tid = threadIdx.x;
    const int bl  = blockIdx.x;
    const int b   = bl / NLEV, l = bl % NLEV;
    int cn = (int)cnt[bl]; if (cn > CANDCAP) cn = CANDCAP;

    for (int i = tid; i < CANDCAP; i += 512) {
        if (i < cn) { key[i] = candV[(size_t)bl*CANDCAP + i]; val[i] = candI[(size_t)bl*CANDCAP + i]; }
        else        { key[i] = -1.0e30f; val[i] = 0u; }
    }
    __syncthreads();
    for (int k = 2; k <= CANDCAP; k <<= 1)
        for (int j = k >> 1; j > 0; j >>= 1) {
            for (int i = tid; i < CANDCAP; i += 512) {
                int ix = i ^ j;
                if (ix > i) {
                    bool up = ((i & k) == 0);       // descending overall
                    float a = key[i], c = key[ix];
                    if (up ? (a < c) : (a > c)) {
                        key[i] = c; key[ix] = a;
                        unsigned int t = val[i]; val[i] = val[ix]; val[ix] = t;
                    }
                }
            }
            __syncthreads();
        }

    const float ims = (float)(*imgsz);
    for (int r = tid; r < TOPN; r += 512) {
        size_t mo = (size_t)b * MPER + (size_t)l * TOPN + r;
        if (r < cn) {
            float w = key[r];
            unsigned int code = val[r];
            int gloc = (int)(code / NCLS);
            int c    = (int)(code % NCLS);
            float lx = locs[2*gloc], ly = locs[2*gloc + 1];
            const float* rg = reg + ((size_t)b * NLOC + gloc) * 4;
            float x1 = lx - rg[0], y1 = ly - rg[1], x2 = lx + rg[2], y2 = ly + rg[3];
            x1 = fminf(fmaxf(x1, 0.f), ims); y1 = fminf(fmaxf(y1, 0.f), ims);
            x2 = fminf(fmaxf(x2, 0.f), ims); y2 = fminf(fmaxf(y2, 0.f), ims);
            mbox[mo*4+0] = x1; mbox[mo*4+1] = y1; mbox[mo*4+2] = x2; mbox[mo*4+3] = y2;
            msc[mo]  = sqrtf(fmaxf(w, 0.f));
            mcls[mo] = (unsigned int)(c + 1);
        } else {
            mbox[mo*4+0] = 0.f; mbox[mo*4+1] = 0.f; mbox[mo*4+2] = 0.f; mbox[mo*4+3] = 0.f;
            msc[mo]  = -1.0f;   // invalid slot (matches NEG semantics)
            mcls[mo] = 0u;
        }
    }
}

// ============================================================
// Kernel 5: per-batch argsort + serial NMS + top-100 post-select
// grid = BATCH blocks x 1024 threads, 140KB dynamic LDS (CDNA5 320KB/WGP)
// ============================================================
__global__ __launch_bounds__(1024) void k_nms(
    const float* __restrict__ mbox, const float* __restrict__ msc,
    const unsigned int* __restrict__ mcls,
    unsigned int* __restrict__ order_ws, float* __restrict__ sscore_ws,
    float* __restrict__ out)
{
    extern __shared__ char smem[];
    __shared__ float s_thr;
    const int tid = threadIdx.x;
    const int b   = blockIdx.x;

    // ---- phase 1: bitonic argsort (descending) over 8192-padded scores ----
    float*        skey = (float*)smem;
    unsigned int* sidx = (unsigned int*)(smem + (size_t)SORTN * 4);
    for (int i = tid; i < SORTN; i += 1024) {
        if (i < MPER) { skey[i] = msc[(size_t)b*MPER + i]; sidx[i] = (unsigned int)i; }
        else          { skey[i] = -3.0e38f;               sidx[i] = 0u; }
    }
    __syncthreads();
    for (int k = 2; k <= SORTN; k <<= 1)
        for (int j = k >> 1; j > 0; j >>= 1) {
            for (int i = tid; i < SORTN; i += 1024) {
                int ix = i ^ j;
                if (ix > i) {
                    bool up = ((i & k) == 0);
                    float a = skey[i], c = skey[ix];
                    if (up ? (a < c) : (a > c)) {
                        skey[i] = c; skey[ix] = a;
                        unsigned int t = sidx[i]; sidx[i] = sidx[ix]; sidx[ix] = t;
                    }
                }
            }
            __syncthreads();
        }
    for (int i = tid; i < MPER; i += 1024) {
        order_ws[(size_t)b*MPAD + i]  = sidx[i];
        sscore_ws[(size_t)b*MPAD + i] = skey[i];
    }
    __threadfence();
    __syncthreads();

    // ---- phase 2: gather sorted boxes into LDS (reuses sort buffer) ----
    float* x1 = (float*)smem;
    float* y1 = x1 + MPAD;
    float* x2 = y1 + MPAD;
    float* y2 = x2 + MPAD;
    float* ar = y2 + MPAD;
    float* sc = ar + MPAD;
    float* kp = sc + MPAD;
    for (int i = tid; i < MPER; i += 1024) {
        unsigned int o = order_ws[(size_t)b*MPAD + i];
        const float* bx = mbox + ((size_t)b*MPER + o) * 4;
        float a0 = bx[0], a1 = bx[1], a2 = bx[2], a3 = bx[3];
        x1[i] = a0; y1[i] = a1; x2[i] = a2; y2[i] = a3;
        ar[i] = (a2 - a0) * (a3 - a1);
        float s = sscore_ws[(size_t)b*MPAD + i];
        sc[i] = s;
        kp[i] = (s >= 0.f) ? 1.f : 0.f;     // valid <=> score >= 0
    }
    __syncthreads();

    // ---- phase 3: serial greedy NMS ----
    for (int i = 0; i < MPER; ++i) {
        if (kp[i] != 0.f) {
            float xi1 = x1[i], yi1 = y1[i], xi2 = x2[i], yi2 = y2[i], ai = ar[i];
            for (int j = i + 1 + tid; j < MPER; j += 1024) {
                if (kp[j] != 0.f) {
                    float bx1 = fmaxf(xi1, x1[j]);
                    float by1 = fmaxf(yi1, y1[j]);
                    float bx2 = fminf(xi2, x2[j]);
                    float by2 = fminf(yi2, y2[j]);
                    float inter = fmaxf(bx2 - bx1, 0.f) * fmaxf(by2 - by1, 0.f);
                    float iou = inter / fmaxf(ai + ar[j] - inter, 1e-9f);
                    if (iou > NMS_THR) kp[j] = 0.f;
                }
            }
        }
        __syncthreads();
    }

    // ---- phase 4: post-select (top-100 among kept; scores sorted desc) ----
    if (tid == 0) {
        int n = 0; float s100 = -3.0e38f;
        for (int i = 0; i < MPER; ++i)
            if (kp[i] != 0.f) { ++n; if (n == POSTN) s100 = sc[i]; }
        s_thr = (n > POSTN) ? s100 : -3.0e38f;
    }
    __syncthreads();
    const float th = s_thr;

    float* obox = out;
    float* osc  = out + (size_t)BATCH * MPER * 4;
    float* ocls = osc + (size_t)BATCH * MPER;
    float* okp  = ocls + (size_t)BATCH * MPER;
    for (int i = tid; i < MPER; i += 1024) {
        bool keep = (kp[i] != 0.f) && (sc[i] >= th) && (sc[i] >= PRE_THR);
        float km = keep ? 1.f : 0.f;
        size_t go = (size_t)b * MPER + i;
        obox[go*4+0] = x1[i] * km; obox[go*4+1] = y1[i] * km;
        obox[go*4+2] = x2[i] * km; obox[go*4+3] = y2[i] * km;
        osc[go] = sc[i] * km;
        unsigned int o = order_ws[(size_t)b*MPAD + i];
        ocls[go] = keep ? (float)mcls[(size_t)b*MPER + o] : 0.f;
        okp[go]  = km;
    }
}

// ============================================================
// Host launcher
// ============================================================
extern "C" void kernel_launch(void* const* d_in, const int* in_sizes, int n_in,
                              void* d_out, int out_size, void* d_ws, size_t ws_size,
                              hipStream_t stream)
{
    const float* cls  = (const float*)d_in[0];   // (16, 21824, 80)
    const float* reg  = (const float*)d_in[1];   // (16, 21824, 4)
    const float* ctr  = (const float*)d_in[2];   // (16, 21824)
    const float* locs = (const float*)d_in[3];   // (21824, 2)
    const int*   imsz = (const int*)d_in[4];     // scalar 1024

    uint8_t* ws = (uint8_t*)d_ws;
    size_t o = 0;
    unsigned int* hist  = (unsigned int*)(ws + o); o += (size_t)BATCH*NLEV*NBIN*4;     // 320KB
    int*          Tbin  = (int*)(ws + o);          o += (size_t)BATCH*NLEV*4;
    float*        tot   = (float*)(ws + o);        o += (size_t)BATCH*NLEV*4;
    unsigned int* cnt   = (unsigned int*)(ws + o); o += (size_t)BATCH*NLEV*4;
    float*        candV = (float*)(ws + o);        o += (size_t)BATCH*NLEV*CANDCAP*4;  // 1.31MB
    unsigned int* candI = (unsigned int*)(ws + o); o += (size_t)BATCH*NLEV*CANDCAP*4;  // 1.31MB
    float*        mbox  = (float*)(ws + o);        o += (size_t)BATCH*MPER*4*4;        // 1.28MB
    float*        mscv  = (float*)(ws + o);        o += (size_t)BATCH*MPER*4;
    unsigned int* mclsv = (unsigned int*)(ws + o); o += (size_t)BATCH*MPER*4;
    unsigned int* ords  = (unsigned int*)(ws + o); o += (size_t)BATCH*MPAD*4;
    float*        ssc   = (float*)(ws + o);        o += (size_t)BATCH*MPAD*4;          // ~5.5MB total

    // Histogram/counters must start at zero every call (graph-replay safe).
    (void)hipMemsetAsync(hist, 0, (size_t)BATCH*NLEV*NBIN*4, stream);
    (void)hipMemsetAsync(cnt,  0, (size_t)BATCH*NLEV*4, stream);

    k_hist     <<<dim3(BATCH*NCHUNK), dim3(256), 0, stream>>>(cls, ctr, hist);
    k_thresh   <<<dim3(BATCH*NLEV),   dim3(64),  0, stream>>>(hist, Tbin, tot);
    k_collect  <<<dim3(BATCH*NCHUNK), dim3(256), 0, stream>>>(cls, ctr, Tbin, cnt, candV, candI);
    k_sortlevel<<<dim3(BATCH*NLEV),   dim3(512), 0, stream>>>(candV, candI, cnt, reg, locs, imsz,
                                                              mbox, mscv, mclsv);
    size_t smem = (size_t)7 * MPAD * 4;   // 140,224B >= 8192*8 sort buffer; fits 320KB WGP LDS
    k_nms      <<<dim3(BATCH),        dim3(1024), smem, stream>>>(mbox, mscv, mclsv, ords, ssc,
                                                                  (float*)d_out);
    (void)in_sizes; (void)n_in; (void)out_size; (void)ws_size;
}